// GatedGCNNet_66374424592521
// MI455X (gfx1250) — compile-verified
//
#include <hip/hip_runtime.h>
#include <hip/hip_bf16.h>

typedef float v2f __attribute__((ext_vector_type(2)));
typedef float v8f __attribute__((ext_vector_type(8)));

#define N_NODES 100000LL
#define N_EDGES 1000000LL
#define HID 64
#define LAYERS 4
#define KMAX 72               // encoder K (max K across all GEMMs)

// ---------------------------------------------------------------------------
// GEMM: C[M,64] = A[M,K] @ B[K,64] + bias, optional ReLU.
// B is staged once per block into LDS, pre-swizzled into exact WMMA fragment
// order: dword ((k4*4 + t)*32 + lane)*2 + which holds
//   which==0 -> B[k4*4 + 2*hi    ][t*16 + r]
//   which==1 -> B[k4*4 + 2*hi + 1][t*16 + r]     (lane = hi*16 + r)
// so the inner loop is one ds_load_b64 per t-tile, landing in an aligned
// VGPR pair that feeds v_wmma_f32_16x16x4_f32 directly (no shuffles), with
// banks {2*lane, 2*lane+1} -> all 64 banks exactly once (conflict-free).
// One wave per 16-row tile; 4 v8f accumulators cover N=64.
// ---------------------------------------------------------------------------
__global__ void gemm_n64_wmma(const float* __restrict__ A,
                              const float* __restrict__ B,
                              const float* __restrict__ bias,
                              float* __restrict__ C,
                              int M, int K, int lda, int relu)
{
    __shared__ float Bs[KMAX * 64];   // 18 KB, fragment-ordered

    // Swizzled fill (gather from L2-hot B, linear LDS stores).
    for (int j = threadIdx.x; j < K * 64; j += blockDim.x) {
        const int which = j & 1;
        const int p     = j >> 1;
        const int ln    = p & 31;
        const int t     = (p >> 5) & 3;
        const int k4    = p >> 7;
        const int row   = k4 * 4 + 2 * (ln >> 4) + which;
        const int col   = t * 16 + (ln & 15);
        Bs[j] = B[row * 64 + col];
    }
    __syncthreads();

    const int lane = threadIdx.x & 31;
    const int wid  = (int)((blockIdx.x * blockDim.x + threadIdx.x) >> 5);
    const int m0   = wid * 16;
    if (m0 >= M) return;                 // wave-uniform: EXEC all-ones for WMMA

    const int hi = lane >> 4;            // half-wave select
    const int r  = lane & 15;            // row (A/C) or col (B/C) within tile

    v8f acc[4];
#pragma unroll
    for (int t = 0; t < 4; ++t) {
        const float bv = bias[t * 16 + r];
#pragma unroll
        for (int g = 0; g < 8; ++g) acc[t][g] = bv;
    }

    const v2f* __restrict__ Bf = (const v2f*)Bs;          // pair-indexed
    const float* __restrict__ Arow = A + (size_t)(m0 + r) * lda;
    const int nk4 = K >> 2;
    for (int k4 = 0; k4 < nk4; ++k4) {
        v2f a;
        a.x = Arow[k4 * 4 + 2 * hi];
        a.y = Arow[k4 * 4 + 2 * hi + 1];
        const int base = (k4 * 4) * 32 + lane;
#pragma unroll
        for (int t = 0; t < 4; ++t) {
            const v2f b = Bf[base + t * 32];              // ds_load_b64
            acc[t] = __builtin_amdgcn_wmma_f32_16x16x4_f32(
                false, a, false, b, (short)0, acc[t], false, false);
        }
    }

#pragma unroll
    for (int t = 0; t < 4; ++t) {
#pragma unroll
        for (int g = 0; g < 8; ++g) {
            float v = acc[t][g];
            if (relu) v = fmaxf(v, 0.0f);
            C[(size_t)(m0 + g + 8 * hi) * 64 + t * 16 + r] = v;
        }
    }
}

// ---------------------------------------------------------------------------
// Edge encoder: e[i,c] = ef[i,0]*W[0,c] + ef[i,1]*W[1,c] + b[c]   (K=2)
// ---------------------------------------------------------------------------
__global__ void edge_encode(const float* __restrict__ ef,
                            const float* __restrict__ W,
                            const float* __restrict__ b,
                            float* __restrict__ e, long long n)
{
    long long t = (long long)blockIdx.x * blockDim.x + threadIdx.x;
    if (t >= n) return;
    const int c = (int)(t & 63);
    const long long i = t >> 6;
    e[t] = ef[i * 2 + 0] * W[c] + ef[i * 2 + 1] * W[64 + c] + b[c];
}

// ---------------------------------------------------------------------------
// Edge gate (grid-stride, stride % 64 == 0 so channel is fixed per thread):
//   e_new += Dh[src] + Eh[dst] (in-place on Ce); sigmoid gate;
//   atomic scatter of gated message/gate onto dst; fused BN partial stats
//   (register accum -> LDS reduce -> 2 atomics per channel per block).
// ---------------------------------------------------------------------------
__global__ void edge_gate(float* __restrict__ en_io,
                          const float* __restrict__ Bh,
                          const float* __restrict__ Dh,
                          const float* __restrict__ Eh,
                          const int* __restrict__ src,
                          const int* __restrict__ dst,
                          float* __restrict__ num,
                          float* __restrict__ den,
                          float* __restrict__ sums, long long n)
{
    __shared__ float ssum[256];
    __shared__ float ssq[256];
    const int c = (int)(threadIdx.x & 63);
    const long long stride = (long long)gridDim.x * blockDim.x;
    float s = 0.0f, q = 0.0f;
    for (long long t = (long long)blockIdx.x * blockDim.x + threadIdx.x;
         t < n; t += stride) {
        const long long i = t >> 6;
        const int sn = src[i];
        const int dn = dst[i];
        float en = en_io[t] + Dh[(size_t)sn * 64 + c] + Eh[(size_t)dn * 64 + c];
        en_io[t] = en;
        s += en;
        q += en * en;
        const float sig = 1.0f / (1.0f + __expf(-en));
        atomicAdd(&num[(size_t)dn * 64 + c], sig * Bh[(size_t)sn * 64 + c]);
        atomicAdd(&den[(size_t)dn * 64 + c], sig);
    }
    ssum[threadIdx.x] = s;
    ssq[threadIdx.x]  = q;
    __syncthreads();
    if (threadIdx.x < 64) {
        s = ssum[c] + ssum[64 + c] + ssum[128 + c] + ssum[192 + c];
        q = ssq[c]  + ssq[64 + c]  + ssq[128 + c]  + ssq[192 + c];
        atomicAdd(&sums[c], s);
        atomicAdd(&sums[64 + c], q);
    }
}

// ---------------------------------------------------------------------------
// Node combine (grid-stride): Ah += num/(den+1e-6); fused BN partial stats.
// ---------------------------------------------------------------------------
__global__ void node_combine(float* __restrict__ Ah,
                             const float* __restrict__ num,
                             const float* __restrict__ den,
                             float* __restrict__ sums, long long n)
{
    __shared__ float ssum[256];
    __shared__ float ssq[256];
    const int c = (int)(threadIdx.x & 63);
    const long long stride = (long long)gridDim.x * blockDim.x;
    float s = 0.0f, q = 0.0f;
    for (long long t = (long long)blockIdx.x * blockDim.x + threadIdx.x;
         t < n; t += stride) {
        const float v = Ah[t] + num[t] / (den[t] + 1e-6f);
        Ah[t] = v;
        s += v;
        q += v * v;
    }
    ssum[threadIdx.x] = s;
    ssq[threadIdx.x]  = q;
    __syncthreads();
    if (threadIdx.x < 64) {
        s = ssum[c] + ssum[64 + c] + ssum[128 + c] + ssum[192 + c];
        q = ssq[c]  + ssq[64 + c]  + ssq[128 + c]  + ssq[192 + c];
        atomicAdd(&sums[c], s);
        atomicAdd(&sums[64 + c], q);
    }
}

// ---------------------------------------------------------------------------
// BN(train, biased var) + affine + ReLU + residual: res += relu(bn(x))
// ---------------------------------------------------------------------------
__global__ void bn_apply(float* __restrict__ res,
                         const float* __restrict__ x,
                         const float* __restrict__ sums,
                         const float* __restrict__ gma,
                         const float* __restrict__ beta, long long rows)
{
    long long t = (long long)blockIdx.x * blockDim.x + threadIdx.x;
    const long long n = rows * 64;
    if (t >= n) return;
    const int c = (int)(t & 63);
    const float invn = 1.0f / (float)rows;
    const float m = sums[c] * invn;
    const float v = sums[64 + c] * invn - m * m;
    const float xn = (x[t] - m) * rsqrtf(v + 1e-5f) * gma[c] + beta[c];
    res[t] += fmaxf(xn, 0.0f);
}

// ---------------------------------------------------------------------------
// Readout tail: out[n,j] = b2[j] + sum_k z[n,k] * W2[k,j]   (64 -> 5)
// ---------------------------------------------------------------------------
__global__ void readout5(const float* __restrict__ z,
                         const float* __restrict__ W2,
                         const float* __restrict__ b2,
                         float* __restrict__ out, int n)
{
    const int t = blockIdx.x * blockDim.x + threadIdx.x;
    if (t >= n * 5) return;
    const int j = t % 5;
    const int row = t / 5;
    const float* __restrict__ zr = z + (size_t)row * 64;
    float acc = b2[j];
#pragma unroll
    for (int k = 0; k < 64; ++k) acc += zr[k] * W2[k * 5 + j];
    out[t] = acc;
}

static inline unsigned cdiv(long long a, long long b) { return (unsigned)((a + b - 1) / b); }

extern "C" void kernel_launch(void* const* d_in, const int* in_sizes, int n_in,
                              void* d_out, int out_size, void* d_ws, size_t ws_size,
                              hipStream_t stream)
{
    (void)in_sizes; (void)n_in; (void)out_size; (void)ws_size;

    // Inputs in setup_inputs() dict order
    const float* h_feats = (const float*)d_in[0];
    const float* e_feats = (const float*)d_in[1];
    const int*   src     = (const int*)d_in[2];
    const int*   dst     = (const int*)d_in[3];
    const float* node_W  = (const float*)d_in[4];
    const float* node_b  = (const float*)d_in[5];
    const float* edge_W  = (const float*)d_in[6];
    const float* edge_b  = (const float*)d_in[7];
    const float* Wa = (const float*)d_in[8];   const float* ba = (const float*)d_in[9];
    const float* Wb = (const float*)d_in[10];  const float* bb = (const float*)d_in[11];
    const float* Wc = (const float*)d_in[12];  const float* bc = (const float*)d_in[13];
    const float* Wd = (const float*)d_in[14];  const float* bd = (const float*)d_in[15];
    const float* We = (const float*)d_in[16];  const float* be = (const float*)d_in[17];
    const float* bn_h_g = (const float*)d_in[18];
    const float* bn_h_b = (const float*)d_in[19];
    const float* bn_e_g = (const float*)d_in[20];
    const float* bn_e_b = (const float*)d_in[21];
    const float* mlp_W1 = (const float*)d_in[22];
    const float* mlp_b1 = (const float*)d_in[23];
    const float* mlp_W2 = (const float*)d_in[24];
    const float* mlp_b2 = (const float*)d_in[25];
    float* out = (float*)d_out;

    // Workspace carve-up (floats)
    const long long NH = N_NODES * HID;    // 6.4M
    const long long EH = N_EDGES * HID;    // 64M
    float* ws   = (float*)d_ws;
    float* h    = ws;            ws += NH;
    float* e    = ws;            ws += EH;
    float* enew = ws;            ws += EH;   // Ce, then e_new in-place
    float* Ah   = ws;            ws += NH;   // also hc, also z
    float* Bh   = ws;            ws += NH;
    float* Dh   = ws;            ws += NH;
    float* Eh   = ws;            ws += NH;
    float* numb = ws;            ws += NH;
    float* denb = ws;            ws += NH;
    float* sums = ws;            ws += 128;

    const int TB = 256;
    const unsigned gemm_blocks_n = cdiv(N_NODES / 16, 8);   // 8 waves/block
    const unsigned gemm_blocks_e = cdiv(N_EDGES / 16, 8);
    const unsigned eh_blocks = cdiv(EH, TB);
    const unsigned nh_blocks = cdiv(NH, TB);

    // Encoders
    gemm_n64_wmma<<<gemm_blocks_n, TB, 0, stream>>>(h_feats, node_W, node_b, h,
                                                    (int)N_NODES, 72, 72, 0);
    edge_encode<<<eh_blocks, TB, 0, stream>>>(e_feats, edge_W, edge_b, e, EH);

    for (int l = 0; l < LAYERS; ++l) {
        const float* Wal = Wa + l * 4096;  const float* bal = ba + l * 64;
        const float* Wbl = Wb + l * 4096;  const float* bbl = bb + l * 64;
        const float* Wcl = Wc + l * 4096;  const float* bcl = bc + l * 64;
        const float* Wdl = Wd + l * 4096;  const float* bdl = bd + l * 64;
        const float* Wel = We + l * 4096;  const float* bel = be + l * 64;

        gemm_n64_wmma<<<gemm_blocks_n, TB, 0, stream>>>(h, Wal, bal, Ah, (int)N_NODES, 64, 64, 0);
        gemm_n64_wmma<<<gemm_blocks_n, TB, 0, stream>>>(h, Wbl, bbl, Bh, (int)N_NODES, 64, 64, 0);
        gemm_n64_wmma<<<gemm_blocks_n, TB, 0, stream>>>(h, Wdl, bdl, Dh, (int)N_NODES, 64, 64, 0);
        gemm_n64_wmma<<<gemm_blocks_n, TB, 0, stream>>>(h, Wel, bel, Eh, (int)N_NODES, 64, 64, 0);
        gemm_n64_wmma<<<gemm_blocks_e, TB, 0, stream>>>(e, Wcl, bcl, enew, (int)N_EDGES, 64, 64, 0);

        // Gather + gate + scatter + fused edge-BN stats
        hipMemsetAsync(numb, 0, NH * sizeof(float), stream);
        hipMemsetAsync(denb, 0, NH * sizeof(float), stream);
        hipMemsetAsync(sums, 0, 128 * sizeof(float), stream);
        edge_gate<<<2048, TB, 0, stream>>>(enew, Bh, Dh, Eh, src, dst, numb, denb, sums, EH);
        bn_apply<<<eh_blocks, TB, 0, stream>>>(e, enew, sums, bn_e_g + l * 64, bn_e_b + l * 64, N_EDGES);

        // Node combine + fused node-BN stats
        hipMemsetAsync(sums, 0, 128 * sizeof(float), stream);
        node_combine<<<1024, TB, 0, stream>>>(Ah, numb, denb, sums, NH);
        bn_apply<<<nh_blocks, TB, 0, stream>>>(h, Ah, sums, bn_h_g + l * 64, bn_h_b + l * 64, N_NODES);
    }

    // MLP head: z = relu(h @ W1 + b1); out = z @ W2 + b2
    gemm_n64_wmma<<<gemm_blocks_n, TB, 0, stream>>>(h, mlp_W1, mlp_b1, Ah, (int)N_NODES, 64, 64, 1);
    readout5<<<cdiv(N_NODES * 5, TB), TB, 0, stream>>>(Ah, mlp_W2, mlp_b2, out, (int)N_NODES);
}